// VarifoldLossTorch_77163382440707
// MI455X (gfx1250) — compile-verified
//
#include <hip/hip_runtime.h>
#include <hip/hip_bf16.h>

typedef __attribute__((ext_vector_type(2))) float v2f;
typedef __attribute__((ext_vector_type(8))) float v8f;

#define BATCH 4
#define NPTS 4096
#define NTILE (NPTS / 16)                       // 256 16-wide tiles per dim
#define WPB 8                                   // waves per block (256 thr)
#define TGROUPS (NTILE / WPB)                   // 32 row-tile groups
#define BLOCKS_PER_TERM (BATCH * TGROUPS)       // 128
#define NBLOCKS (3 * BLOCKS_PER_TERM)           // 384
#define WAVES_TOTAL (NBLOCKS * WPB)             // 3072 partial sums

__global__ __launch_bounds__(256) void varifold_wmma_kernel(
    const float* __restrict__ xyz1, const float* __restrict__ xyz2,
    const float* __restrict__ nor1, const float* __restrict__ nor2,
    float* __restrict__ partial)
{
    const int lane = threadIdx.x & 31;
    const int wave = threadIdx.x >> 5;
    const int half = lane >> 4;       // K-half this lane carries (0: K0..1, 1: K2..3)
    const int sub  = lane & 15;       // M (for A) / N (for B) index within tile

    const int term = blockIdx.x / BLOCKS_PER_TERM;
    const int rem  = blockIdx.x % BLOCKS_PER_TERM;
    const int b    = rem / TGROUPS;
    const int tg   = rem % TGROUPS;
    const int ti   = tg * WPB + wave; // this wave's 16-row tile

    const float *Xa, *Na, *Xb, *Nb;
    float w;
    if (term == 0)      { Xa = xyz1; Na = nor1; Xb = xyz1; Nb = nor1; w =  1.0f; }
    else if (term == 1) { Xa = xyz2; Na = nor2; Xb = xyz2; Nb = nor2; w =  1.0f; }
    else                { Xa = xyz1; Na = nor1; Xb = xyz2; Nb = nor2; w = -2.0f; }

    const size_t base = (size_t)b * NPTS * 3;

    // ---- A fragments (loaded once per wave): row m = ti*16 + sub ----
    const int m = ti * 16 + sub;
    const float ax = Xa[base + m * 3 + 0];
    const float ay = Xa[base + m * 3 + 1];
    const float az = Xa[base + m * 3 + 2];
    const float rn = ax * ax + ay * ay + az * az;   // ||x_i||^2 folded into K=3 slot
    const float nx = Na[base + m * 3 + 0];
    const float ny = Na[base + m * 3 + 1];
    const float nz = Na[base + m * 3 + 2];

    v2f aS, aD;
    if (half) { aS.x = az; aS.y = rn;   aD.x = nz; aD.y = 0.0f; }
    else      { aS.x = ax; aS.y = ay;   aD.x = nx; aD.y = ny;   }

    float acc = 0.0f;

    for (int tj = 0; tj < NTILE; ++tj) {
        // ---- B fragments: column n = tj*16 + sub ----
        const int n = tj * 16 + sub;
        const float bx = Xb[base + n * 3 + 0];
        const float by = Xb[base + n * 3 + 1];
        const float bz = Xb[base + n * 3 + 2];
        const float cn = bx * bx + by * by + bz * bz;  // ||x_j||^2, per-lane (N = lane%16)
        const float mx = Nb[base + n * 3 + 0];
        const float my = Nb[base + n * 3 + 1];
        const float mz = Nb[base + n * 3 + 2];

        v2f bS, bD;
        if (half) { bS.x = -2.0f * bz; bS.y = 1.0f;        bD.x = mz; bD.y = 0.0f; }
        else      { bS.x = -2.0f * bx; bS.y = -2.0f * by;  bD.x = mx; bD.y = my;   }

        // Prefetch next column tile (global_prefetch_b8)
        if (tj + 1 < NTILE) {
            __builtin_prefetch(&Xb[base + (n + 16) * 3], 0, 1);
            __builtin_prefetch(&Nb[base + (n + 16) * 3], 0, 1);
        }

        // cS[m][n] = ||x_i||^2 - 2 x_i . x_j   ;   cD[m][n] = n_i . n_j
        v8f cS = {};
        v8f cD = {};
        cS = __builtin_amdgcn_wmma_f32_16x16x4_f32(false, aS, false, bS,
                                                   (short)0, cS, false, false);
        cD = __builtin_amdgcn_wmma_f32_16x16x4_f32(false, aD, false, bD,
                                                   (short)0, cD, false, false);

        // Elementwise: acc += exp(-S) * D^2 ; S = cS + ||x_j||^2
        #pragma unroll
        for (int r = 0; r < 8; ++r) {
            const float S = cS[r] + cn;
            const float d = cD[r];
            acc += __expf(-S) * d * d;
        }
    }

    // Wave32 reduction (deterministic in-wave tree)
    #pragma unroll
    for (int off = 16; off > 0; off >>= 1)
        acc += __shfl_xor(acc, off, 32);

    if (lane == 0)
        partial[blockIdx.x * WPB + wave] = acc * w;
}

// Deterministic fixed-order final reduction of the 3072 per-wave partials.
__global__ __launch_bounds__(256) void varifold_reduce_kernel(
    const float* __restrict__ partial, float* __restrict__ out)
{
    __shared__ float sm[256];
    float acc = 0.0f;
    for (int i = threadIdx.x; i < WAVES_TOTAL; i += 256)
        acc += partial[i];
    sm[threadIdx.x] = acc;
    __syncthreads();
    for (int s = 128; s > 0; s >>= 1) {
        if ((int)threadIdx.x < s) sm[threadIdx.x] += sm[threadIdx.x + s];
        __syncthreads();
    }
    if (threadIdx.x == 0) out[0] = sm[0];
}

extern "C" void kernel_launch(void* const* d_in, const int* in_sizes, int n_in,
                              void* d_out, int out_size, void* d_ws, size_t ws_size,
                              hipStream_t stream) {
    const float* xyz1 = (const float*)d_in[0];
    const float* xyz2 = (const float*)d_in[1];
    const float* nor1 = (const float*)d_in[2];
    const float* nor2 = (const float*)d_in[3];
    float* partial = (float*)d_ws;   // 3072 floats of scratch

    varifold_wmma_kernel<<<NBLOCKS, 256, 0, stream>>>(xyz1, xyz2, nor1, nor2, partial);
    varifold_reduce_kernel<<<1, 256, 0, stream>>>(partial, (float*)d_out);
}